// PeakDetectionLoss_44092134261358
// MI455X (gfx1250) — compile-verified
//
#include <hip/hip_runtime.h>
#include <hip/hip_bf16.h>
#include <stdint.h>

// ---------------------------------------------------------------------------
// PeakDetectionLoss for MI455X (gfx1250).
//
// Math reduction: normalization is affine+monotone -> peak mask invariant;
// amplitude filter reduces to  x_raw > (mean_peak_raw + mean_raw)/2.
// => two streaming passes, no std needed.  ~160MB traffic => ~7us @ 23.3TB/s,
// pass 2 L2-resident (80MB < 192MB L2).  All reductions are fixed-order
// (deterministic); no float atomics.
// ---------------------------------------------------------------------------

#define NSIG   5
#define SIGS   (2 * NSIG)           // 0..4 = rppg rows, 5..9 = ppg rows
#define LSIG   2097152
#define WWIN   11
#define HALO   5
#define CHUNK  2048
#define CB     (LSIG / CHUNK)       // 1024 chunks per signal
#define NSMEM  (CHUNK + 2 * HALO)   // 2058 floats staged per block

typedef float        __attribute__((ext_vector_type(2)))  v2f;
typedef float        __attribute__((ext_vector_type(8)))  v8f;
typedef unsigned int __attribute__((ext_vector_type(4)))  v4u;
typedef int          __attribute__((ext_vector_type(4)))  v4i;
typedef int          __attribute__((ext_vector_type(8)))  v8i;

struct Rec { int first; int last; int cnt; float inv; };

// ---------------------------------------------------------------------------
// TDM: one async DMA per block stages [j0,j1) of the chunk(+halo) into LDS.
// D# built per CDNA5 ISA ch.8 (group0: count/lds_addr/global_addr/type=2;
// group1: data_size=4B, tensor_dim0 = tile_dim0 = element count, 1 row).
// ---------------------------------------------------------------------------
#if defined(__has_builtin)
#if __has_builtin(__builtin_amdgcn_tensor_load_to_lds) && __has_builtin(__builtin_amdgcn_s_wait_tensorcnt)
#define USE_TDM 1
#endif
#endif

#ifdef USE_TDM
__device__ __forceinline__ void tdm_load_1d(uint32_t lds_byte_addr,
                                            const float* gptr, uint32_t nelem) {
  uint64_t ga = (uint64_t)(uintptr_t)gptr;
  v4u g0;
  g0.x = 1u;                                              // count=1 valid D#
  g0.y = lds_byte_addr;                                   // lds_addr [63:32]
  g0.z = (uint32_t)(ga & 0xFFFFFFFFu);                    // global_addr lo
  g0.w = (uint32_t)((ga >> 32) & 0x01FFFFFFu) | (2u << 30); // addr hi | type=2
  v8i g1;
  g1[0] = (int)(2u << 16);                                // data_size=2 (4B)
  g1[1] = (int)((nelem & 0xFFFFu) << 16);                 // tensor_dim0 [15:0]
  g1[2] = (int)(((nelem >> 16) & 0xFFFFu) | (1u << 16));  // dim0 hi | tensor_dim1=1
  g1[3] = (int)((nelem & 0xFFFFu) << 16);                 // tile_dim0
  g1[4] = 0;                                              // tile_dim1/2 unused
  g1[5] = (int)nelem;                                     // tensor_dim0_stride lo
  g1[6] = 0;
  g1[7] = 0;
  v4i z4 = {0, 0, 0, 0};
#if __has_include(<hip/amd_detail/amd_gfx1250_TDM.h>)
  v8i z8 = {0, 0, 0, 0, 0, 0, 0, 0};
  __builtin_amdgcn_tensor_load_to_lds(g0, g1, z4, z4, z8, 0);   // clang-23 form
#else
  __builtin_amdgcn_tensor_load_to_lds(g0, g1, z4, z4, 0);       // ROCm 7.2 form
#endif
}
#endif

// Stage chunk cb of a signal into smem[0..NSMEM): smem[j] == x[cb*CHUNK-HALO+j],
// with -inf outside the signal (matches reduce_window's -inf padding).
__device__ __forceinline__ void stage_chunk(float* smem, const float* sigbase, int cb) {
  const int tid = threadIdx.x;
  const int j0 = (cb == 0) ? HALO : 0;
  const int j1 = (cb == CB - 1) ? (CHUNK + HALO) : NSMEM;
  const long lo = (long)cb * CHUNK - HALO;      // global index of smem[0]
  if (tid < 16) {                               // -inf halo at signal edges
    if (tid < j0) smem[tid] = -__builtin_inff();
    int jt = NSMEM - 1 - tid;
    if (jt >= j1) smem[jt] = -__builtin_inff();
  }
#ifdef USE_TDM
  if (tid < 32) {                               // wave 0 issues one DMA
    tdm_load_1d((uint32_t)(uintptr_t)(&smem[j0]), sigbase + (lo + j0),
                (uint32_t)(j1 - j0));
    __builtin_amdgcn_s_wait_tensorcnt(0);
  }
#else
  for (int j = j0 + tid; j < j1; j += 256) smem[j] = sigbase[lo + j];
#endif
  __syncthreads();
}

__device__ __forceinline__ const float* sig_base(const float* rppg, const float* ppg,
                                                 int sig) {
  return (sig < NSIG) ? (rppg + (size_t)sig * LSIG)
                      : (ppg + (size_t)(sig - NSIG) * LSIG);
}

// ---------------------------------------------------------------------------
// K1: per-chunk {sum(x), sum(x at peaks), #peaks}.  8 elems/thread; window-max
// from an 18-float register window.
// ---------------------------------------------------------------------------
__global__ __launch_bounds__(256) void k1_stats(const float* __restrict__ rppg,
                                                const float* __restrict__ ppg,
                                                float* __restrict__ partials) {
  __shared__ float smem[NSMEM];
  __shared__ float red[3][256];
  const int sig = blockIdx.y, cb = blockIdx.x, tid = threadIdx.x;
  stage_chunk(smem, sig_base(rppg, ppg, sig), cb);

  float w[18];
  const int lb = tid * 8;                       // smem index of w[0]
#pragma unroll
  for (int i = 0; i < 18; ++i) w[i] = smem[lb + i];

  float sumx = 0.f, pksum = 0.f, pkcnt = 0.f;
#pragma unroll
  for (int i = 0; i < 8; ++i) {
    float x = w[i + 5];
    float m = w[i];
#pragma unroll
    for (int k = 1; k < WWIN; ++k) m = fmaxf(m, w[i + k]);
    sumx += x;
    if (x == m) { pkcnt += 1.f; pksum += x; }
  }
  red[0][tid] = sumx; red[1][tid] = pksum; red[2][tid] = pkcnt;
  __syncthreads();
  for (int s = 128; s > 0; s >>= 1) {           // deterministic tree reduce
    if (tid < s) {
      red[0][tid] += red[0][tid + s];
      red[1][tid] += red[1][tid + s];
      red[2][tid] += red[2][tid + s];
    }
    __syncthreads();
  }
  if (tid == 0) {
    float* o = partials + ((size_t)sig * CB + cb) * 3;
    o[0] = red[0][0]; o[1] = red[1][0]; o[2] = red[2][0];
  }
}

// ---------------------------------------------------------------------------
// K2: reduce 1024 chunk partials per signal -> threshold (mp+mu)/2.
// WMMA trick: with A = all-ones, D += A*B adds per-column sums of B into D;
// Sum_n D[0][n] == sum of ALL 64 B elements, independent of B's lane layout.
// Row 0 of D lives in D-VGPR0, lanes 0..15 -> small LDS finish.
// ---------------------------------------------------------------------------
#if defined(__has_builtin)
#if __has_builtin(__builtin_amdgcn_wmma_f32_16x16x4_f32)
#define USE_WMMA_REDUCE 1
#endif
#endif

__global__ __launch_bounds__(32) void k2_reduce(const float* __restrict__ partials,
                                                float* __restrict__ thresh) {
  const int sig = blockIdx.x, lane = threadIdx.x;   // one wave32 per signal
  const float* p = partials + (size_t)sig * CB * 3;
#ifdef USE_WMMA_REDUCE
  __shared__ float tmp[3][32];
  const v2f a = {1.0f, 1.0f};                       // all-ones A (16x4)
  v8f acc0 = {}, acc1 = {}, acc2 = {};
  for (int it = 0; it < CB / 64; ++it) {            // 64 values / wave / step
    const int i0 = it * 64 + lane, i1 = i0 + 32;
    v2f b0 = {p[i0 * 3 + 0], p[i1 * 3 + 0]};
    v2f b1 = {p[i0 * 3 + 1], p[i1 * 3 + 1]};
    v2f b2 = {p[i0 * 3 + 2], p[i1 * 3 + 2]};
    acc0 = __builtin_amdgcn_wmma_f32_16x16x4_f32(false, a, false, b0, (short)0, acc0, false, false);
    acc1 = __builtin_amdgcn_wmma_f32_16x16x4_f32(false, a, false, b1, (short)0, acc1, false, false);
    acc2 = __builtin_amdgcn_wmma_f32_16x16x4_f32(false, a, false, b2, (short)0, acc2, false, false);
  }
  tmp[0][lane] = acc0[0]; tmp[1][lane] = acc1[0]; tmp[2][lane] = acc2[0];
  __syncthreads();
  if (lane == 0) {
    float s0 = 0.f, s1 = 0.f, s2 = 0.f;
    for (int i = 0; i < 16; ++i) { s0 += tmp[0][i]; s1 += tmp[1][i]; s2 += tmp[2][i]; }
    const float mu = s0 / (float)LSIG;
    const float mp = s1 / s2;
    thresh[sig] = 0.5f * (mp + mu);
  }
#else
  __shared__ float tmp[3][32];
  float s0 = 0.f, s1 = 0.f, s2 = 0.f;
  for (int c = lane; c < CB; c += 32) {
    s0 += p[c * 3 + 0]; s1 += p[c * 3 + 1]; s2 += p[c * 3 + 2];
  }
  tmp[0][lane] = s0; tmp[1][lane] = s1; tmp[2][lane] = s2;
  __syncthreads();
  if (lane == 0) {
    s0 = s1 = s2 = 0.f;
    for (int i = 0; i < 32; ++i) { s0 += tmp[0][i]; s1 += tmp[1][i]; s2 += tmp[2][i]; }
    thresh[sig] = 0.5f * (s1 / s2 + s0 / (float)LSIG);
  }
#endif
}

// ---------------------------------------------------------------------------
// K3: filtered peaks per chunk.  Peaks are >=6 apart (two peaks within the
// window dominate each other), so <=2 per 8-elem thread, <=342 per chunk.
// Block scan orders them; record = {first, last, count, sum 1/gap (internal)}.
// ---------------------------------------------------------------------------
__global__ __launch_bounds__(256) void k3_peaks(const float* __restrict__ rppg,
                                                const float* __restrict__ ppg,
                                                const float* __restrict__ thresh,
                                                Rec* __restrict__ recs) {
  __shared__ float smem[NSMEM];
  __shared__ int scan[256];
  __shared__ int peaks[384];
  __shared__ float red[256];
  const int sig = blockIdx.y, cb = blockIdx.x, tid = threadIdx.x;
  stage_chunk(smem, sig_base(rppg, ppg, sig), cb);
  const float t = thresh[sig];

  float w[18];
  const int lb = tid * 8;
#pragma unroll
  for (int i = 0; i < 18; ++i) w[i] = smem[lb + i];

  int cnt = 0, loc0 = 0, loc1 = 0;
#pragma unroll
  for (int i = 0; i < 8; ++i) {
    float x = w[i + 5];
    float m = w[i];
#pragma unroll
    for (int k = 1; k < WWIN; ++k) m = fmaxf(m, w[i + k]);
    if (x == m && x > t) {
      int gi = cb * CHUNK + tid * 8 + i;
      if (cnt == 0) loc0 = gi; else if (cnt == 1) loc1 = gi;
      if (cnt < 2) ++cnt;                      // spacing>=6 => never exceeds 2
    }
  }
  scan[tid] = cnt;
  __syncthreads();
  for (int off = 1; off < 256; off <<= 1) {    // inclusive Hillis-Steele scan
    int v = scan[tid];
    int ad = (tid >= off) ? scan[tid - off] : 0;
    __syncthreads();
    scan[tid] = v + ad;
    __syncthreads();
  }
  const int total = scan[255];
  const int base = scan[tid] - cnt;
  if (cnt > 0) peaks[base] = loc0;
  if (cnt > 1) peaks[base + 1] = loc1;
  __syncthreads();

  float inv = 0.f;
  for (int i = tid + 1; i < total; i += 256)
    inv += 1.0f / (float)(peaks[i] - peaks[i - 1]);
  red[tid] = inv;
  __syncthreads();
  for (int s = 128; s > 0; s >>= 1) {
    if (tid < s) red[tid] += red[tid + s];
    __syncthreads();
  }
  if (tid == 0) {
    Rec r;
    r.cnt   = total;
    r.first = total ? peaks[0] : -1;
    r.last  = total ? peaks[total - 1] : -1;
    r.inv   = red[0];
    recs[(size_t)sig * CB + cb] = r;
  }
}

// ---------------------------------------------------------------------------
// K4: stitch chunk records sequentially per signal (adds the cross-chunk
// gaps), compute HR, then the final score.  10 cheap sequential scans.
// ---------------------------------------------------------------------------
__global__ __launch_bounds__(32) void k4_final(const Rec* __restrict__ recs,
                                               const int* __restrict__ fsp,
                                               float* __restrict__ out) {
  __shared__ float hr[16];
  const int tid = threadIdx.x;
  if (tid < SIGS) {
    const Rec* r = recs + (size_t)tid * CB;
    float inv = 0.f; int valid = 0; int prev = -1;
    for (int c = 0; c < CB; ++c) {
      Rec rc = r[c];
      if (rc.cnt > 0) {
        if (prev >= 0) { inv += 1.0f / (float)(rc.first - prev); ++valid; }
        inv += rc.inv;
        valid += rc.cnt - 1;
        prev = rc.last;
      }
    }
    hr[tid] = 60.0f * (float)fsp[0] * inv / (float)valid;
  }
  __syncthreads();
  if (tid == 0) {
    float s = 0.f;
    for (int n = 0; n < NSIG; ++n)
      s += fabsf(hr[NSIG + n] - hr[n]) / hr[NSIG + n];   // ppg_hr = hr[5..9]
    out[0] = s / (float)NSIG;
  }
}

// ---------------------------------------------------------------------------
extern "C" void kernel_launch(void* const* d_in, const int* in_sizes, int n_in,
                              void* d_out, int out_size, void* d_ws, size_t ws_size,
                              hipStream_t stream) {
  const float* rppg = (const float*)d_in[0];
  const float* ppg  = (const float*)d_in[1];
  const int*   fs   = (const int*)d_in[2];
  float* out = (float*)d_out;

  // Workspace layout (all regions fully written before being read):
  //   partials : SIGS*CB*3 f32  = 122880 B
  //   thresh   : 16 f32 (10 used)
  //   recs     : SIGS*CB Rec    = 163840 B          (total ~287 KB)
  float* partials = (float*)d_ws;
  float* thresh   = partials + (size_t)SIGS * CB * 3;
  Rec*   recs     = (Rec*)(thresh + 16);

  dim3 grid(CB, SIGS);
  k1_stats<<<grid, 256, 0, stream>>>(rppg, ppg, partials);
  k2_reduce<<<SIGS, 32, 0, stream>>>(partials, thresh);
  k3_peaks<<<grid, 256, 0, stream>>>(rppg, ppg, thresh, recs);
  k4_final<<<1, 32, 0, stream>>>(recs, fs, out);
}